// sp_reduce_10479720202785
// MI455X (gfx1250) — compile-verified
//
#include <hip/hip_runtime.h>

// Streaming stencil + global reduction. Memory-bound (~2.5 flop/byte) =>
// WMMA deliberately unused. CDNA5-specific paths:
//   - global_load_async_to_lds_b128 (+ s_wait_asynccnt) for tile staging
//   - global_prefetch_b8 (L2 prefetch of next tile)
//   - ds_* LDS reductions, split wait counters

#define TPB   256
#define VEC   4
#define TILE  (TPB * VEC)      // 1024 elements per block
#define WIN   (TILE + 8)       // staged window: [base-4, base+1028), 1032 floats
#define NSLOT (WIN / 4)        // 258 float4 slots

__device__ __forceinline__ float load_guard(const float* __restrict__ f,
                                            int idx, int n) {
    return (idx >= 0 && idx < n) ? f[idx] : 0.0f;
}

// gfx1250 async DMA: 16B global -> LDS, no VGPR roundtrip.
// Low 32 bits of a flat LDS address = wave-relative LDS byte offset,
// which is exactly what the instruction's VDST expects.
__device__ __forceinline__ void async_cp16(const float* gp, float* dst) {
    unsigned lds_addr = (unsigned)(unsigned long long)(void*)dst;
    asm volatile("global_load_async_to_lds_b128 %0, %1, off"
                 :
                 : "v"(lds_addr), "v"(gp)
                 : "memory");
}

// Stage [base-4, base+1028) of zero-padded f into LDS.
// Interior blocks (all but first/last): unguarded async fast path.
__device__ __forceinline__ void stage_tile(const float* __restrict__ f,
                                           int n, int base, float* smem) {
    const int tid = threadIdx.x;
    if (base >= 4 && base + TILE + 4 <= n) {          // uniform fast path
        const float* gwin = f + (base - 4);           // window start
        async_cp16(gwin + 4 * tid, smem + 4 * tid);
        if (tid < NSLOT - TPB)                        // slots 256, 257
            async_cp16(gwin + 4 * (TPB + tid), smem + 4 * (TPB + tid));
        if (base + 2 * TILE + 4 <= n)                 // L2 prefetch next tile
            __builtin_prefetch(gwin + 4 * tid + TILE, 0, 1);
    } else {                                          // edge blocks only
        for (int v = tid; v < NSLOT; v += TPB) {
            int g = base - 4 + 4 * v;
            float4 val;
            val.x = load_guard(f, g + 0, n);
            val.y = load_guard(f, g + 1, n);
            val.z = load_guard(f, g + 2, n);
            val.w = load_guard(f, g + 3, n);
            *reinterpret_cast<float4*>(smem + 4 * v) = val;
        }
    }
    // Wait for this wave's async DMA writes to land in LDS, then sync waves.
    asm volatile("s_wait_asynccnt 0x0" ::: "memory");
    __syncthreads();
}

// Pass 1: per-block partial sum of ((f[j-2] - f[j+3]) / 5)^2 over j in [0, n-1).
__global__ void diffsq_partial_kernel(const float* __restrict__ f,
                                      float* __restrict__ partials, int n) {
    __shared__ __align__(16) float smem[WIN];
    __shared__ float red[TPB];
    const int base = (int)blockIdx.x * TILE;
    stage_tile(f, n, base, smem);

    const int t4 = threadIdx.x * VEC;
    float acc = 0.0f;
#pragma unroll
    for (int u = 0; u < VEC; ++u) {
        int j = base + t4 + u;
        if (j < n - 1) {
            // f[j-2] -> smem[t4+u+2], f[j+3] -> smem[t4+u+7]
            float d = (smem[t4 + u + 2] - smem[t4 + u + 7]) * 0.2f;
            acc += d * d;
        }
    }
    red[threadIdx.x] = acc;
    __syncthreads();
#pragma unroll
    for (int s = TPB / 2; s > 0; s >>= 1) {
        if ((int)threadIdx.x < s) red[threadIdx.x] += red[threadIdx.x + s];
        __syncthreads();
    }
    if (threadIdx.x == 0) partials[blockIdx.x] = red[0];
}

// Pass 2: single-block deterministic reduction (fixed order, fixed tree
// -> bit-identical sum on every replay; no float atomics anywhere).
__global__ void reduce_partials_kernel(const float* __restrict__ partials,
                                       float* __restrict__ sum_out, int nblk) {
    __shared__ float red[TPB];
    float acc = 0.0f;
    for (int i = threadIdx.x; i < nblk; i += TPB) acc += partials[i];
    red[threadIdx.x] = acc;
    __syncthreads();
#pragma unroll
    for (int s = TPB / 2; s > 0; s >>= 1) {
        if ((int)threadIdx.x < s) red[threadIdx.x] += red[threadIdx.x + s];
        __syncthreads();
    }
    if (threadIdx.x == 0) sum_out[0] = red[0];
}

// Pass 3: out[i] = f[i] * (1 - change_i)^5,
//   change_i = (i==0) ? 0 : ((f[i-3]-f[i+2])/5)^2 / S
__global__ void scale_kernel(const float* __restrict__ f,
                             const float* __restrict__ sum_in,
                             float* __restrict__ out, int n) {
    __shared__ __align__(16) float smem[WIN];
    const int base = (int)blockIdx.x * TILE;
    stage_tile(f, n, base, smem);

    const float invS = 1.0f / sum_in[0];   // uniform scalar load, L2-resident
    const int t4 = threadIdx.x * VEC;
    float o[VEC];
#pragma unroll
    for (int u = 0; u < VEC; ++u) {
        int i = base + t4 + u;
        float c = 0.0f;
        if (i != 0) {
            // f[i-3] -> smem[t4+u+1], f[i+2] -> smem[t4+u+6]
            float d = (smem[t4 + u + 1] - smem[t4 + u + 6]) * 0.2f;
            c = d * d * invS;
        }
        float t  = 1.0f - c;
        float t2 = t * t;
        o[u] = smem[t4 + u + 4] * t2 * t2 * t;   // f[i] * t^5
    }
    const int i0 = base + t4;
    if (i0 + VEC <= n) {
        float4 v = make_float4(o[0], o[1], o[2], o[3]);
        *reinterpret_cast<float4*>(out + i0) = v;      // b128 store
    } else {
#pragma unroll
        for (int u = 0; u < VEC; ++u)
            if (i0 + u < n) out[i0 + u] = o[u];
    }
}

extern "C" void kernel_launch(void* const* d_in, const int* in_sizes, int n_in,
                              void* d_out, int out_size, void* d_ws, size_t ws_size,
                              hipStream_t stream) {
    const float* f   = (const float*)d_in[0];
    float*       out = (float*)d_out;
    const int    n   = in_sizes[0];
    const int  nblk  = (n + TILE - 1) / TILE;   // 16384 for n = 2^24

    float* partials = (float*)d_ws;             // nblk floats
    float* sum_out  = partials + nblk;          // 1 float

    diffsq_partial_kernel<<<nblk, TPB, 0, stream>>>(f, partials, n);
    reduce_partials_kernel<<<1, TPB, 0, stream>>>(partials, sum_out, nblk);
    scale_kernel<<<nblk, TPB, 0, stream>>>(f, sum_out, out, n);
}